// ForwardDeformer_64733747086015
// MI455X (gfx1250) — compile-verified
//
#include <hip/hip_runtime.h>
#include <math.h>

typedef float v2f __attribute__((ext_vector_type(2)));
typedef float v4f __attribute__((ext_vector_type(4)));
typedef float v8f __attribute__((ext_vector_type(8)));

#define NJ 25
#define NI 14
#define VD 32
#define VH 128
#define VW 128
#define JSTRIDE (VD*VH*VW)             /* 524288 elements between joints */
#define SOFTK 28.853900817779268f      /* 20 * log2(e) */
#define BLOCK 256
#define NWAVE (BLOCK/32)
#define WROW 28                        /* K=25 padded to 28 (7 k-tiles of 4) */
#define MROW 20                        /* padded D row: conflict-free, 16B aligned */
#define NEG_INF_F -10000.0f

struct __attribute__((packed, aligned(4))) F2 { v2f v; };
__device__ __forceinline__ v2f ld2(const float* p) { return ((const F2*)p)->v; }

// ---------------------------------------------------------------------------
// Kernel 1: affine 4x4 inverse of tfs[init_bones[i]] (last row is [0,0,0,1]).
// Writes 14 x (3x4) row-major inverse blocks into workspace.
// ---------------------------------------------------------------------------
__global__ void tinv_kernel(const float* __restrict__ tfs,
                            const int* __restrict__ bones,
                            float* __restrict__ tinv, int nb)
{
    int i = threadIdx.x;
    if (i >= nb) return;
    const float* M = tfs + bones[i] * 16;
    float a00=M[0],a01=M[1],a02=M[2],t0=M[3];
    float a10=M[4],a11=M[5],a12=M[6],t1=M[7];
    float a20=M[8],a21=M[9],a22=M[10],t2=M[11];
    float c00=a11*a22-a12*a21, c01=a02*a21-a01*a22, c02=a01*a12-a02*a11;
    float c10=a12*a20-a10*a22, c11=a00*a22-a02*a20, c12=a02*a10-a00*a12;
    float c20=a10*a21-a11*a20, c21=a01*a20-a00*a21, c22=a00*a11-a01*a10;
    float det = a00*c00 + a01*c10 + a02*c20;
    float r = 1.0f / det;
    c00*=r; c01*=r; c02*=r; c10*=r; c11*=r; c12*=r; c20*=r; c21*=r; c22*=r;
    float b0 = -(c00*t0 + c01*t1 + c02*t2);
    float b1 = -(c10*t0 + c11*t1 + c12*t2);
    float b2 = -(c20*t0 + c21*t1 + c22*t2);
    float* o = tinv + i*12;
    o[0]=c00; o[1]=c01; o[2]=c02;  o[3]=b0;
    o[4]=c10; o[5]=c11; o[6]=c12;  o[7]=b1;
    o[8]=c20; o[9]=c21; o[10]=c22; o[11]=b2;
}

// ---------------------------------------------------------------------------
// Kernel 2: lane-per-point deformer. Per init bone: inverse-warp, 25-channel
// trilinear gather, softmax, then a wave-cooperative (32x25)x(25x12) GEMM via
// v_wmma_f32_16x16x4_f32 (2 row tiles x 7 K-slices) to blend the 25 joint
// transforms; err / mask / argmax aggregation in registers.
// ---------------------------------------------------------------------------
__global__ __launch_bounds__(BLOCK)
void deformer_kernel(const float* __restrict__ x,
                     const float* __restrict__ tfs,
                     const float* __restrict__ vox,
                     const float* __restrict__ tinv,
                     float* __restrict__ out, int N)
{
    __shared__ __align__(16) float sTi[NI*12];               // inverse init-bone tfs
    __shared__ __align__(16) float sT [NJ*12];               // tfs[:, :3, :] flattened (B matrix)
    __shared__ __align__(16) float sW [NWAVE*32*WROW];       // softmax numerators (A matrix rows)
    __shared__ __align__(16) float sM [NWAVE*2*16*MROW];     // D scatter/gather buffer

    const int t = threadIdx.x;
    for (int u = t; u < NI*12; u += BLOCK) sTi[u] = tinv[u];
    for (int u = t; u < NJ*12; u += BLOCK) sT[u] = tfs[(u/12)*16 + (u%12)];
    __syncthreads();

    const int lane  = t & 31;
    const int wave  = t >> 5;
    const int nlo   = lane & 15;
    const int khalf = (lane >> 4) * 2;

    // Constant B operand tiles: B[k][n] = sT[k*12 + n] (k = joint, n = 12 transform
    // entries), zero-padded to K=28, N=16. f32 WMMA 4x16 B layout (mirrors the
    // documented 16x4 A layout): v0 = {K0 | lanes0-15, K2 | lanes16-31}, v1 = {K1, K3}.
    v2f Bk[7];
#pragma unroll
    for (int kt = 0; kt < 7; ++kt) {
        int k0 = kt*4 + khalf;
        v2f b;
        b.x = (k0   < NJ && nlo < 12) ? sT[ k0   *12 + nlo] : 0.0f;
        b.y = (k0+1 < NJ && nlo < 12) ? sT[(k0+1)*12 + nlo] : 0.0f;
        Bk[kt] = b;
    }

    const int n  = blockIdx.x * BLOCK + t;
    const int np = (n < N) ? n : (N - 1);
    const float px = x[np*3+0], py = x[np*3+1], pz = x[np*3+2];

    float best_d = -3.0e38f, best_m = 0.0f;
    float bw0 = 0.0f, bw1 = 0.0f, bw2 = 0.0f;
    float bf[15];
#pragma unroll
    for (int k = 0; k < 15; ++k) bf[k] = 0.0f;

    float*       wrow  = sW + (wave*32 + lane)*WROW;
    const float* arow0 = sW + wave*32*WROW + nlo*WROW + khalf;
    const float* arow1 = arow0 + 16*WROW;
    float*       mst   = sM + wave*2*16*MROW;
    float*       st0   = mst + (8*(lane>>4))*MROW + nlo;     // D[m = r + 8*half][n]
    float*       st1   = st0 + 16*MROW;
    const float* mrd   = mst + (lane>>4)*16*MROW + nlo*MROW; // my point's 12-value row

    float s[28];
    s[25] = 0.0f; s[26] = 0.0f; s[27] = 0.0f;                // K padding for the GEMM

    for (int i = 0; i < NI; ++i) {
        // xc = (Tinv_i @ [p,1])[:3]
        const v4f* T4 = (const v4f*)(sTi + i*12);
        v4f Tr0 = T4[0], Tr1 = T4[1], Tr2 = T4[2];
        float xcx = fmaf(Tr0.x,px, fmaf(Tr0.y,py, fmaf(Tr0.z,pz, Tr0.w)));
        float xcy = fmaf(Tr1.x,px, fmaf(Tr1.y,py, fmaf(Tr1.z,pz, Tr1.w)));
        float xcz = fmaf(Tr2.x,px, fmaf(Tr2.y,py, fmaf(Tr2.z,pz, Tr2.w)));

        // grid coords, align_corners + border clamp; pair base clamped to size-2
        // so the contiguous b64 corner-pair load is always in bounds.
        float cx = fminf(fmaxf((xcx+1.0f)*63.5f, 0.0f), 127.0f);
        float cy = fminf(fmaxf((xcy+1.0f)*63.5f, 0.0f), 127.0f);
        float cz = fminf(fmaxf((xcz+1.0f)*15.5f, 0.0f),  31.0f);
        float xbf = fminf(floorf(cx), 126.0f);
        float ybf = fminf(floorf(cy), 126.0f);
        float zbf = fminf(floorf(cz),  30.0f);
        float fx = cx - xbf, fy = cy - ybf, fz = cz - zbf;
        float gx = 1.0f-fx,  gy = 1.0f-fy,  gz = 1.0f-fz;
        float wy00 = gy*gz, wy10 = fy*gz, wy01 = gy*fz, wy11 = fy*fz;
        float w000 = gx*wy00, w100 = fx*wy00;
        float w010 = gx*wy10, w110 = fx*wy10;
        float w001 = gx*wy01, w101 = fx*wy01;
        float w011 = gx*wy11, w111 = fx*wy11;
        int xb = (int)xbf, yb = (int)ybf, zb = (int)zbf;
        const float* vp = vox + ((zb*VH + yb)*VW + xb);

        // 25-channel trilinear gather: 4 corner-pair b64 loads + 8 FMA per joint
        float mx = -3.0e38f;
#pragma unroll
        for (int j = 0; j < NJ; ++j) {
            const float* q = vp + j*JSTRIDE;
            v2f a = ld2(q);
            v2f b = ld2(q + VW);
            v2f c = ld2(q + VH*VW);
            v2f d = ld2(q + VH*VW + VW);
            float v =            w000*a.x;
            v = fmaf(w100, a.y, v);
            v = fmaf(w010, b.x, v);
            v = fmaf(w110, b.y, v);
            v = fmaf(w001, c.x, v);
            v = fmaf(w101, c.y, v);
            v = fmaf(w011, d.x, v);
            v = fmaf(w111, d.y, v);
            s[j] = v;
            mx = fmaxf(mx, v);
        }

        // softmax numerators (shift-invariant; exp2 maps to v_exp_f32)
        float sum = 0.0f;
#pragma unroll
        for (int j = 0; j < NJ; ++j) {
            float e = exp2f((s[j] - mx) * SOFTK);
            s[j] = e;
            sum += e;
        }

        // stage A rows (this wave only; DS ops are in-order per wave -> no barrier)
        v4f* wr4 = (v4f*)wrow;
#pragma unroll
        for (int c = 0; c < 7; ++c) {
            v4f qv; qv.x = s[4*c]; qv.y = s[4*c+1]; qv.z = s[4*c+2]; qv.w = s[4*c+3];
            wr4[c] = qv;
        }

        // D(32x12) = W(32x25) @ T(25x12): 2 row tiles x 7 K-slices of WMMA f32 16x16x4
        v8f acc0 = {0,0,0,0,0,0,0,0};
        v8f acc1 = {0,0,0,0,0,0,0,0};
#pragma unroll
        for (int kt = 0; kt < 7; ++kt) {
            v2f a0 = *(const v2f*)(arow0 + kt*4);
            v2f a1 = *(const v2f*)(arow1 + kt*4);
            acc0 = __builtin_amdgcn_wmma_f32_16x16x4_f32(false, a0, false, Bk[kt],
                                                         (short)0, acc0, false, false);
            acc1 = __builtin_amdgcn_wmma_f32_16x16x4_f32(false, a1, false, Bk[kt],
                                                         (short)0, acc1, false, false);
        }

        // scatter D (documented C/D layout), gather this lane's blended transform M
#pragma unroll
        for (int r = 0; r < 8; ++r) { st0[r*MROW] = acc0[r]; st1[r*MROW] = acc1[r]; }
        const v4f* mr4 = (const v4f*)mrd;
        v4f m0 = mr4[0], m1 = mr4[1], m2 = mr4[2];

        float rs  = 1.0f / sum;
        float wxp = fmaf(m0.x,xcx, fmaf(m0.y,xcy, fmaf(m0.z,xcz, m0.w))) * rs;
        float wyp = fmaf(m1.x,xcx, fmaf(m1.y,xcy, fmaf(m1.z,xcz, m1.w))) * rs;
        float wzp = fmaf(m2.x,xcx, fmaf(m2.y,xcy, fmaf(m2.z,xcz, m2.w))) * rs;
        float dx = wxp - px, dy = wyp - py, dz = wzp - pz;
        float err = sqrtf(fmaf(dx,dx, fmaf(dy,dy, dz*dz)));
        bool  msk  = err < 0.1f;
        float dens = msk ? -err : NEG_INF_F;
        if (dens > best_d) {
            best_d = dens;
            best_m = msk ? 1.0f : 0.0f;
            bw0 = xcx; bw1 = xcy; bw2 = xcz;
#pragma unroll
            for (int k = 0; k < 15; ++k) bf[k] = s[k] * rs;
        }
    }

    if (n < N) {
        out[n] = best_d;                                  // d_agg   [N]
        float* fo = out + N + n*15;                       // f_agg   [N,15]
#pragma unroll
        for (int k = 0; k < 15; ++k) fo[k] = bf[k];
        out[(size_t)16*N + n] = best_m;                   // m_agg   [N]
        float* wo = out + (size_t)17*N + n*3;             // w_agg   [N,3]
        wo[0] = bw0; wo[1] = bw1; wo[2] = bw2;
    }
}

// ---------------------------------------------------------------------------
extern "C" void kernel_launch(void* const* d_in, const int* in_sizes, int n_in,
                              void* d_out, int out_size, void* d_ws, size_t ws_size,
                              hipStream_t stream)
{
    (void)n_in; (void)out_size; (void)ws_size;
    const float* x     = (const float*)d_in[0];
    const float* tfs   = (const float*)d_in[1];
    const float* vox   = (const float*)d_in[2];
    const int*   bones = (const int*)d_in[3];
    float*       out   = (float*)d_out;
    float*       tinv  = (float*)d_ws;                    // 14*12 floats = 672 B

    int N = in_sizes[0] / 3;

    tinv_kernel<<<1, 32, 0, stream>>>(tfs, bones, tinv, NI);
    int grid = (N + BLOCK - 1) / BLOCK;
    deformer_kernel<<<grid, BLOCK, 0, stream>>>(x, tfs, vox, tinv, out, N);
}